// GatedDeltaUnit_26525718020280
// MI455X (gfx1250) — compile-verified
//
#include <hip/hip_runtime.h>

// ---------------------------------------------------------------------------
// GatedDeltaUnit for MI455X (gfx1250, wave32).
// fp32 WMMA (v_wmma_f32_16x16x4_f32) for all GEMMs (exact f32 semantics).
// 128x64 macro-tiles, branch-free b128 global<->LDS staging (LDS stride 36
// floats: conflict-free fragment reads + 16B-aligned rows), non-temporal
// hints on all streaming traffic so the GEMM working set stays L2-resident.
// ---------------------------------------------------------------------------

typedef __attribute__((ext_vector_type(2))) float v2f;
typedef __attribute__((ext_vector_type(4))) float v4f;
typedef __attribute__((ext_vector_type(8))) float v8f;

#define B_   4
#define S_   4096
#define DIN  1024
#define DH   2048
#define DR   256
#define DC   (DIN + DH)        // 3072, combined width
#define M_   (B_ * S_)         // 16384 rows

#define KT   32                // K tile per LDS stage
#define LDT  36                // padded LDS row stride (floats): 36*l mod 64
                               // distinct for l=0..15 -> conflict-free, and
                               // rows stay 16B aligned for b128 staging.
#define MT   128               // M tile (k_main)
#define NTL  64                // N tile

__device__ __forceinline__ float silu_f(float x) { return x / (1.0f + __expf(-x)); }
__device__ __forceinline__ float sigm_f(float x) { return 1.0f / (1.0f + __expf(-x)); }

// ---------------------------------------------------------------------------
// Kernel 0: per-batch initial-hidden projections (tiny).
//   dh[b][r] = sum_c h0[b][c] * W_delta[r][DIN + c]          (B*DR outputs)
//   gh[b][h] = sum_c h0[b][c] * W_gate [h][DIN + c]          (B*DH outputs)
// ---------------------------------------------------------------------------
__global__ __launch_bounds__(256) void k_h0proj(
    const float* __restrict__ hidden,
    const float* __restrict__ Wd,
    const float* __restrict__ Wg,
    float* __restrict__ dh,
    float* __restrict__ gh)
{
    const int wid  = (int)((blockIdx.x * blockDim.x + threadIdx.x) >> 5);
    const int lane = (int)(threadIdx.x & 31);

    float sum = 0.0f;
    if (wid < B_ * DR) {
        const int b = wid / DR, r = wid % DR;
        const float* __restrict__ w  = Wd + (size_t)r * DC + DIN;
        const float* __restrict__ h0 = hidden + (size_t)b * DH;
        for (int c = lane; c < DH; c += 32) sum += h0[c] * w[c];
        for (int off = 16; off > 0; off >>= 1) sum += __shfl_xor(sum, off, 32);
        if (lane == 0) dh[wid] = sum;
    } else if (wid < B_ * DR + B_ * DH) {
        const int id = wid - B_ * DR;
        const int b = id / DH, h = id % DH;
        const float* __restrict__ w  = Wg + (size_t)h * DC + DIN;
        const float* __restrict__ h0 = hidden + (size_t)b * DH;
        for (int c = lane; c < DH; c += 32) sum += h0[c] * w[c];
        for (int off = 16; off > 0; off >>= 1) sum += __shfl_xor(sum, off, 32);
        if (lane == 0) gh[id] = sum;
    }
}

// ---------------------------------------------------------------------------
// Kernel 1: dl[m][r] = silu( x[m][:] . W_delta[r][:DIN] + dh[b(m)][r] )
// 64x64 tile, 8 waves (4M x 2N), wave = 16x32 output (two 16x16 subtiles).
// ---------------------------------------------------------------------------
__global__ __launch_bounds__(256) void k_delta_latent(
    const float* __restrict__ x,
    const float* __restrict__ Wd,
    const float* __restrict__ dh,
    float* __restrict__ dl)
{
    __shared__ float lx[64 * LDT];
    __shared__ float lw[64 * LDT];

    const int m0   = (int)blockIdx.x * 64;
    const int n0   = (int)blockIdx.y * 64;
    const int tid  = (int)threadIdx.x;
    const int wave = tid >> 5;
    const int lane = tid & 31;
    const int wm   = wave & 3;
    const int wn   = wave >> 2;
    const int half = lane >> 4;      // lanes 0-15: K+0/K+1; lanes 16-31: K+2/K+3
    const int l    = lane & 15;
    const int R    = wm * 16;
    const int Cn   = wn * 32;

    // branch-free staging coordinates: thread owns (row r8 + 32*rr, cols c4..c4+3)
    const int r8 = tid >> 3;         // 0..31
    const int c4 = (tid & 7) * 4;    // 0,4,...,28

    v8f acc0 = {}; v8f acc1 = {};

    const float* __restrict__ pa  = &lx[(R + l) * LDT + 2 * half];
    const float* __restrict__ pb0 = &lw[(Cn + l) * LDT + 2 * half];
    const float* __restrict__ pb1 = pb0 + 16 * LDT;

    for (int k0 = 0; k0 < DIN; k0 += KT) {
#pragma unroll
        for (int rr = 0; rr < 2; ++rr)
            *(v4f*)&lx[(r8 + rr * 32) * LDT + c4] =
                *(const v4f*)&x[(size_t)(m0 + r8 + rr * 32) * DIN + (k0 + c4)];
#pragma unroll
        for (int rr = 0; rr < 2; ++rr)
            *(v4f*)&lw[(r8 + rr * 32) * LDT + c4] =
                *(const v4f*)&Wd[(size_t)(n0 + r8 + rr * 32) * DC + (k0 + c4)];
        if (k0 + KT < DIN) {   // uniform branch, no exec divergence
            __builtin_prefetch(&x [(size_t)(m0 + r8) * DIN + (k0 + KT + c4)], 0, 0);
            __builtin_prefetch(&Wd[(size_t)(n0 + r8) * DC  + (k0 + KT + c4)], 0, 0);
        }
        __syncthreads();
#pragma unroll
        for (int kk = 0; kk < KT; kk += 4) {
            v2f a, b0, b1;
            a.x  = pa [kk];     a.y  = pa [kk + 1];
            b0.x = pb0[kk];     b0.y = pb0[kk + 1];
            b1.x = pb1[kk];     b1.y = pb1[kk + 1];
            acc0 = __builtin_amdgcn_wmma_f32_16x16x4_f32(false, a, false, b0,
                                                         (short)0, acc0, false, false);
            acc1 = __builtin_amdgcn_wmma_f32_16x16x4_f32(false, a, false, b1,
                                                         (short)0, acc1, false, false);
        }
        __syncthreads();
    }

    const int b = m0 >> 12;                 // m0 / S_, tiles never cross batches
#pragma unroll
    for (int i = 0; i < 8; ++i) {
        const int m = m0 + R + i + 8 * half;
        int n = n0 + Cn + l;
        dl[(size_t)m * DR + n] = silu_f(acc0[i] + dh[b * DR + n]);
        n += 16;
        dl[(size_t)m * DR + n] = silu_f(acc1[i] + dh[b * DR + n]);
    }
}

// ---------------------------------------------------------------------------
// Kernel 2 (fused main), 128x64 tile, 8 waves (4M x 2N), wave = 32x32:
//   v  = silu(x.Wv^T + bv)
//   g  = sigmoid(x.Wg^T + gh[b] + bg)
//   d  = dl.We^T + be
//   a  = 1 - g            -> ws_a   (non-temporal)
//   bb = g * (v + d)      -> d_out  (non-temporal; consumed in-place by scan)
// ---------------------------------------------------------------------------
__global__ __launch_bounds__(256) void k_main(
    const float* __restrict__ x,
    const float* __restrict__ Wv, const float* __restrict__ bv,
    const float* __restrict__ Wg, const float* __restrict__ bg,
    const float* __restrict__ We, const float* __restrict__ be,
    const float* __restrict__ gh,
    const float* __restrict__ dl,
    float* __restrict__ a_out,
    float* __restrict__ b_out)
{
    __shared__ float lx [MT * LDT];   // 18 KB
    __shared__ float lw0[64 * LDT];   // 9 KB
    __shared__ float lw1[64 * LDT];   // 9 KB

    const int m0   = (int)blockIdx.x * MT;
    const int n0   = (int)blockIdx.y * NTL;
    const int tid  = (int)threadIdx.x;
    const int wave = tid >> 5;
    const int lane = tid & 31;
    const int wm   = wave & 3;        // 4 waves over M
    const int wn   = wave >> 2;       // 2 waves over N
    const int half = lane >> 4;
    const int l    = lane & 15;
    const int R    = wm * 32;         // wave covers rows R..R+31 (2 subtiles)
    const int Cn   = wn * 32;         // wave covers cols Cn..Cn+31 (2 subtiles)

    const int r8 = tid >> 3;          // 0..31
    const int c4 = (tid & 7) * 4;     // 0,4,...,28

    v8f av[2][2] = {};                // [mi][ni] value accumulators
    v8f ag[2][2] = {};                // gate
    v8f ad[2][2] = {};                // delta

    const float* __restrict__ pa0  = &lx [(R + l) * LDT + 2 * half];
    const float* __restrict__ pa1  = pa0 + 16 * LDT;
    const float* __restrict__ pb00 = &lw0[(Cn + l) * LDT + 2 * half];
    const float* __restrict__ pb01 = pb00 + 16 * LDT;
    const float* __restrict__ pb10 = &lw1[(Cn + l) * LDT + 2 * half];
    const float* __restrict__ pb11 = pb10 + 16 * LDT;

    // ---- phase 1: K over DIN, A = x tile (128x32), B = Wv / Wg tiles ----
    for (int k0 = 0; k0 < DIN; k0 += KT) {
#pragma unroll
        for (int rr = 0; rr < 4; ++rr)     // x: 128 rows
            *(v4f*)&lx[(r8 + rr * 32) * LDT + c4] =
                *(const v4f*)&x[(size_t)(m0 + r8 + rr * 32) * DIN + (k0 + c4)];
#pragma unroll
        for (int rr = 0; rr < 2; ++rr)     // Wv: 64 rows
            *(v4f*)&lw0[(r8 + rr * 32) * LDT + c4] =
                *(const v4f*)&Wv[(size_t)(n0 + r8 + rr * 32) * DIN + (k0 + c4)];
#pragma unroll
        for (int rr = 0; rr < 2; ++rr)     // Wg: 64 rows
            *(v4f*)&lw1[(r8 + rr * 32) * LDT + c4] =
                *(const v4f*)&Wg[(size_t)(n0 + r8 + rr * 32) * DC + (k0 + c4)];
        if (k0 + KT < DIN) {   // uniform branch
            __builtin_prefetch(&x [(size_t)(m0 + r8) * DIN + (k0 + KT + c4)], 0, 0);
            __builtin_prefetch(&Wv[(size_t)(n0 + r8) * DIN + (k0 + KT + c4)], 0, 0);
            __builtin_prefetch(&Wg[(size_t)(n0 + r8) * DC  + (k0 + KT + c4)], 0, 0);
        }
        __syncthreads();
#pragma unroll
        for (int kk = 0; kk < KT; kk += 4) {
            v2f a0, a1, b0, b1, c0, c1;
            a0.x = pa0 [kk];  a0.y = pa0 [kk + 1];
            a1.x = pa1 [kk];  a1.y = pa1 [kk + 1];
            b0.x = pb00[kk];  b0.y = pb00[kk + 1];
            b1.x = pb01[kk];  b1.y = pb01[kk + 1];
            c0.x = pb10[kk];  c0.y = pb10[kk + 1];
            c1.x = pb11[kk];  c1.y = pb11[kk + 1];
            av[0][0] = __builtin_amdgcn_wmma_f32_16x16x4_f32(false, a0, false, b0,
                                                (short)0, av[0][0], false, false);
            av[0][1] = __builtin_amdgcn_wmma_f32_16x16x4_f32(false, a0, false, b1,
                                                (short)0, av[0][1], false, false);
            av[1][0] = __builtin_amdgcn_wmma_f32_16x16x4_f32(false, a1, false, b0,
                                                (short)0, av[1][0], false, false);
            av[1][1] = __builtin_amdgcn_wmma_f32_16x16x4_f32(false, a1, false, b1,
                                                (short)0, av[1][1], false, false);
            ag[0][0] = __builtin_amdgcn_wmma_f32_16x16x4_f32(false, a0, false, c0,
                                                (short)0, ag[0][0], false, false);
            ag[0][1] = __builtin_amdgcn_wmma_f32_16x16x4_f32(false, a0, false, c1,
                                                (short)0, ag[0][1], false, false);
            ag[1][0] = __builtin_amdgcn_wmma_f32_16x16x4_f32(false, a1, false, c0,
                                                (short)0, ag[1][0], false, false);
            ag[1][1] = __builtin_amdgcn_wmma_f32_16x16x4_f32(false, a1, false, c1,
                                                (short)0, ag[1][1], false, false);
        }
        __syncthreads();
    }

    // ---- phase 2: K over DR, A = dl tile (128x32), B = We tile (lw0) ----
    for (int k0 = 0; k0 < DR; k0 += KT) {
#pragma unroll
        for (int rr = 0; rr < 4; ++rr)     // dl: 128 rows
            *(v4f*)&lx[(r8 + rr * 32) * LDT + c4] =
                *(const v4f*)&dl[(size_t)(m0 + r8 + rr * 32) * DR + (k0 + c4)];
#pragma unroll
        for (int rr = 0; rr < 2; ++rr)     // We: 64 rows
            *(v4f*)&lw0[(r8 + rr * 32) * LDT + c4] =
                *(const v4f*)&We[(size_t)(n0 + r8 + rr * 32) * DR + (k0 + c4)];
        __syncthreads();
#pragma unroll
        for (int kk = 0; kk < KT; kk += 4) {
            v2f a0, a1, b0, b1;
            a0.x = pa0 [kk];  a0.y = pa0 [kk + 1];
            a1.x = pa1 [kk];  a1.y = pa1 [kk + 1];
            b0.x = pb00[kk];  b0.y = pb00[kk + 1];
            b1.x = pb01[kk];  b1.y = pb01[kk + 1];
            ad[0][0] = __builtin_amdgcn_wmma_f32_16x16x4_f32(false, a0, false, b0,
                                                (short)0, ad[0][0], false, false);
            ad[0][1] = __builtin_amdgcn_wmma_f32_16x16x4_f32(false, a0, false, b1,
                                                (short)0, ad[0][1], false, false);
            ad[1][0] = __builtin_amdgcn_wmma_f32_16x16x4_f32(false, a1, false, b0,
                                                (short)0, ad[1][0], false, false);
            ad[1][1] = __builtin_amdgcn_wmma_f32_16x16x4_f32(false, a1, false, b1,
                                                (short)0, ad[1][1], false, false);
        }
        __syncthreads();
    }

    // ---- epilogue: activations, a = 1-g, b = g*(v+d); NT streaming stores ----
    const int b = m0 >> 12;
#pragma unroll
    for (int mi = 0; mi < 2; ++mi)
#pragma unroll
    for (int ni = 0; ni < 2; ++ni)
#pragma unroll
    for (int i = 0; i < 8; ++i) {
        const int m = m0 + R + mi * 16 + i + 8 * half;
        const int n = n0 + Cn + ni * 16 + l;
        const float vv = silu_f(av[mi][ni][i] + bv[n]);
        const float gg = sigm_f(ag[mi][ni][i] + gh[b * DH + n] + bg[n]);
        const float dd = ad[mi][ni][i] + be[n];
        __builtin_nontemporal_store(1.0f - gg,      &a_out[(size_t)m * DH + n]);
        __builtin_nontemporal_store(gg * (vv + dd), &b_out[(size_t)m * DH + n]);
    }
}

// ---------------------------------------------------------------------------
// Kernel 3: sequential linear recurrence, one lane per (b, h) channel.
//   h_t = a_t * h_{t-1} + b_t, h_prev init = hidden.
// Unrolled x4 so 8 independent loads issue ahead of the dependent fma chain.
// a reads are last-use; h stream is write-once -> non-temporal.
// ---------------------------------------------------------------------------
__global__ __launch_bounds__(256) void k_scan(
    const float* __restrict__ hidden,
    const float* __restrict__ a_in,
    float* __restrict__ out)
{
    const int ch = (int)(blockIdx.x * blockDim.x + threadIdx.x);
    if (ch >= B_ * DH) return;
    const int b = ch / DH;
    const int h = ch % DH;

    float hp = hidden[ch];
    const float* __restrict__ ap = a_in + (size_t)b * S_ * DH + h;
    float* __restrict__ op       = out  + (size_t)b * S_ * DH + h;

    for (int s = 0; s < S_; s += 4) {
        const float a0 = __builtin_nontemporal_load(ap + (size_t)(s + 0) * DH);
        const float a1 = __builtin_nontemporal_load(ap + (size_t)(s + 1) * DH);
        const float a2 = __builtin_nontemporal_load(ap + (size_t)(s + 2) * DH);
        const float a3 = __builtin_nontemporal_load(ap + (size_t)(s + 3) * DH);
        const float b0 = __builtin_nontemporal_load(op + (size_t)(s + 0) * DH);
        const float b1 = __builtin_nontemporal_load(op + (size_t)(s + 1) * DH);
        const float b2 = __builtin_nontemporal_load(op + (size_t)(s + 2) * DH);
        const float b3 = __builtin_nontemporal_load(op + (size_t)(s + 3) * DH);
        hp = fmaf(a0, hp, b0);
        __builtin_nontemporal_store(hp, op + (size_t)(s + 0) * DH);
        hp = fmaf(a1, hp, b1);
        __builtin_nontemporal_store(hp, op + (size_t)(s + 1) * DH);
        hp = fmaf(a2, hp, b2);
        __builtin_nontemporal_store(hp, op + (size_t)(s + 2) * DH);
        hp = fmaf(a3, hp, b3);
        __builtin_nontemporal_store(hp, op + (size_t)(s + 3) * DH);
    }
    out[(size_t)B_ * S_ * DH + ch] = hp;   // final hidden
}

// ---------------------------------------------------------------------------
// Launch
// ---------------------------------------------------------------------------
extern "C" void kernel_launch(void* const* d_in, const int* in_sizes, int n_in,
                              void* d_out, int out_size, void* d_ws, size_t ws_size,
                              hipStream_t stream)
{
    (void)in_sizes; (void)n_in; (void)out_size; (void)ws_size;

    const float* x   = (const float*)d_in[0];
    const float* hid = (const float*)d_in[1];
    const float* Wd  = (const float*)d_in[2];
    const float* We  = (const float*)d_in[3];
    const float* be  = (const float*)d_in[4];
    const float* Wg  = (const float*)d_in[5];
    const float* bg  = (const float*)d_in[6];
    const float* Wv  = (const float*)d_in[7];
    const float* bv  = (const float*)d_in[8];

    float* out = (float*)d_out;
    float* ws  = (float*)d_ws;

    float* ws_a  = ws;                               // M_*DH  floats (134 MB)
    float* ws_dl = ws_a  + (size_t)M_ * DH;          // M_*DR  floats (16 MB)
    float* ws_dh = ws_dl + (size_t)M_ * DR;          // B_*DR
    float* ws_gh = ws_dh + (size_t)B_ * DR;          // B_*DH

    // 0) h0 projections: (B*DR + B*DH) waves = 9216 -> 1152 blocks of 8 waves
    k_h0proj<<<1152, 256, 0, stream>>>(hid, Wd, Wg, ws_dh, ws_gh);

    // 1) dl = silu(x . Wd_x^T + dh)
    dim3 gA(M_ / 64, DR / 64);
    k_delta_latent<<<gA, 256, 0, stream>>>(x, Wd, ws_dh, ws_dl);

    // 2) fused v/g/delta GEMMs + epilogue -> a (ws), b (d_out)
    dim3 gB(M_ / MT, DH / NTL);
    k_main<<<gB, 256, 0, stream>>>(x, Wv, bv, Wg, bg, We, be,
                                   ws_gh, ws_dl, ws_a, out);

    // 3) recurrence scan, in place on d_out; appends final hidden
    k_scan<<<(B_ * DH) / 256, 256, 0, stream>>>(hid, ws_a, out);
}